// SparseEncoder_27487790695252
// MI455X (gfx1250) — compile-verified
//
#include <hip/hip_runtime.h>
#include <hip/hip_bf16.h>

typedef __attribute__((ext_vector_type(2))) float v2f;
typedef __attribute__((ext_vector_type(8))) float v8f;

#define IN_F   128
#define LATENT 64
#define CLAMP_MIN -10.0f
#define CLAMP_MAX 3.0f

// ---------------------------------------------------------------------------
// K0: seed output with biases. d_out layout: [mu (N*64) | logvar (N*64)].
// ---------------------------------------------------------------------------
__global__ void init_out(float* __restrict__ out,
                         const float* __restrict__ b_mu,
                         const float* __restrict__ b_lv,
                         long n_nodes) {
    long t = (long)blockIdx.x * blockDim.x + threadIdx.x;
    long half = n_nodes * LATENT;
    if (t < half) {
        out[t] = b_mu[t & (LATENT - 1)];
    } else if (t < 2 * half) {
        out[t] = b_lv[t & (LATENT - 1)];
    }
}

// ---------------------------------------------------------------------------
// K1: h[N,128] = x[N,128] @ [W_mu | W_lv]  using V_WMMA_F32_16X16X4_F32.
// One wave per 16-row M-tile; 8 accumulators cover all 128 output columns.
// A frag (16x4 f32): lane<16 holds x[m=lane][k0], x[m][k0+1];
//                    lane>=16 holds x[m=lane-16][k0+2], x[m][k0+3]  (contig f2)
// B frag (4x16 f32): VGPR0 = W[k0+2*(lane/16)][n], VGPR1 = W[k0+1+2*(lane/16)][n]
// C/D (16x16 f32):   VGPR r -> row = r + 8*(lane/16), col = lane%16
// ---------------------------------------------------------------------------
__global__ void __launch_bounds__(256)
gemm_xw(const float* __restrict__ x,
        const float* __restrict__ Wmu,
        const float* __restrict__ Wlv,
        float* __restrict__ h,
        int m_tiles) {
    const int lane = threadIdx.x & 31;
    const int wave = threadIdx.x >> 5;
    const int tile = blockIdx.x * 8 + wave;
    if (tile >= m_tiles) return;          // uniform per wave; EXEC all-1s inside

    const int half = lane >> 4;           // 0 or 1
    const int l15  = lane & 15;

    const size_t rowA = (size_t)(tile * 16 + l15) * IN_F;

    v8f acc[8];
    #pragma unroll
    for (int nt = 0; nt < 8; ++nt) acc[nt] = (v8f){0,0,0,0,0,0,0,0};

    #pragma unroll 4
    for (int k0 = 0; k0 < IN_F; k0 += 4) {
        // A fragment: contiguous float2 from x
        v2f a = *(const v2f*)(x + rowA + k0 + 2 * half);
        const int kk = k0 + 2 * half;
        #pragma unroll
        for (int nt = 0; nt < 8; ++nt) {
            const float* W = (nt < 4) ? Wmu : Wlv;
            const int col = (nt & 3) * 16 + l15;
            v2f b;
            b.x = W[(size_t)kk * LATENT + col];
            b.y = W[(size_t)(kk + 1) * LATENT + col];
            acc[nt] = __builtin_amdgcn_wmma_f32_16x16x4_f32(
                /*neg_a=*/false, a, /*neg_b=*/false, b,
                /*c_mod=*/(short)0, acc[nt],
                /*reuse_a=*/false, /*reuse_b=*/false);
        }
    }

    // Store h tile (row-major [N,128])
    #pragma unroll
    for (int nt = 0; nt < 8; ++nt) {
        #pragma unroll
        for (int r = 0; r < 8; ++r) {
            const int row = r + 8 * half;
            const int col = nt * 16 + l15;
            h[(size_t)(tile * 16 + row) * IN_F + col] = acc[nt][r];
        }
    }
}

// Scalar fallback for remainder rows (m_start..n_nodes); not hit when N%16==0.
__global__ void gemm_xw_tail(const float* __restrict__ x,
                             const float* __restrict__ Wmu,
                             const float* __restrict__ Wlv,
                             float* __restrict__ h,
                             int m_start, int n_nodes) {
    long t = (long)blockIdx.x * blockDim.x + threadIdx.x;
    long rows = (long)(n_nodes - m_start);
    if (t >= rows * IN_F) return;
    int row = m_start + (int)(t >> 7);
    int col = (int)(t & 127);
    const float* W = (col < LATENT) ? Wmu : Wlv;
    int wc = col & (LATENT - 1);
    float s = 0.0f;
    for (int k = 0; k < IN_F; ++k)
        s += x[(size_t)row * IN_F + k] * W[(size_t)k * LATENT + wc];
    h[(size_t)row * IN_F + col] = s;
}

// ---------------------------------------------------------------------------
// K2: edge scatter in latent space. One wave per edge; each lane handles 4
// contiguous latent features (lanes 0-15 -> mu half, 16-31 -> logvar half).
// ---------------------------------------------------------------------------
__global__ void __launch_bounds__(256)
spmm_scatter(const int* __restrict__ rows,
             const int* __restrict__ cols,
             const float* __restrict__ vals,
             const float* __restrict__ h,
             float* __restrict__ out,
             int n_edges, long n_nodes) {
    const int e    = (int)(((long)blockIdx.x * blockDim.x + threadIdx.x) >> 5);
    const int lane = threadIdx.x & 31;
    if (e >= n_edges) return;

    const int   r = rows[e];
    const int   c = cols[e];
    const float v = vals[e];

    const float4 hv = *(const float4*)(h + (size_t)c * IN_F + lane * 4);

    float* base = (lane < 16)
        ? (out + (size_t)r * LATENT + lane * 4)
        : (out + n_nodes * LATENT + (size_t)r * LATENT + (lane - 16) * 4);

    __hip_atomic_fetch_add(base + 0, v * hv.x, __ATOMIC_RELAXED, __HIP_MEMORY_SCOPE_AGENT);
    __hip_atomic_fetch_add(base + 1, v * hv.y, __ATOMIC_RELAXED, __HIP_MEMORY_SCOPE_AGENT);
    __hip_atomic_fetch_add(base + 2, v * hv.z, __ATOMIC_RELAXED, __HIP_MEMORY_SCOPE_AGENT);
    __hip_atomic_fetch_add(base + 3, v * hv.w, __ATOMIC_RELAXED, __HIP_MEMORY_SCOPE_AGENT);
}

// ---------------------------------------------------------------------------
// K3: clamp logvar half in place (bias already folded in by init_out).
// ---------------------------------------------------------------------------
__global__ void clamp_lv(float* __restrict__ lv, long n) {
    long t = (long)blockIdx.x * blockDim.x + threadIdx.x;
    if (t < n) {
        float vv = lv[t];
        vv = vv < CLAMP_MIN ? CLAMP_MIN : (vv > CLAMP_MAX ? CLAMP_MAX : vv);
        lv[t] = vv;
    }
}

// ---------------------------------------------------------------------------
extern "C" void kernel_launch(void* const* d_in, const int* in_sizes, int n_in,
                              void* d_out, int out_size, void* d_ws, size_t ws_size,
                              hipStream_t stream) {
    const float* x    = (const float*)d_in[0];
    const int*   rows = (const int*)  d_in[1];
    const int*   cols = (const int*)  d_in[2];
    const float* vals = (const float*)d_in[3];
    const float* Wmu  = (const float*)d_in[4];
    const float* bmu  = (const float*)d_in[5];
    const float* Wlv  = (const float*)d_in[6];
    const float* blv  = (const float*)d_in[7];

    float* out = (float*)d_out;
    float* h   = (float*)d_ws;                       // N * 128 floats

    const long n_nodes = in_sizes[0] / IN_F;         // 100000
    const int  n_edges = in_sizes[1];                // 1600000

    // K0: seed output with biases
    {
        long total = 2 * n_nodes * LATENT;
        int blocks = (int)((total + 255) / 256);
        init_out<<<blocks, 256, 0, stream>>>(out, bmu, blv, n_nodes);
    }

    // K1: h = x @ [Wmu | Wlv]   (WMMA f32 16x16x4)
    {
        int m_tiles = (int)(n_nodes / 16);           // 6250 full tiles
        int blocks  = (m_tiles + 7) / 8;             // 8 waves per block
        gemm_xw<<<blocks, 256, 0, stream>>>(x, Wmu, Wlv, h, m_tiles);
        int m_start = m_tiles * 16;
        if (m_start < n_nodes) {
            long rem = (n_nodes - m_start) * (long)IN_F;
            gemm_xw_tail<<<(int)((rem + 255) / 256), 256, 0, stream>>>(
                x, Wmu, Wlv, h, m_start, (int)n_nodes);
        }
    }

    // K2: edge scatter (atomic f32 adds, L2-resident accumulator)
    {
        int blocks = (n_edges + 7) / 8;              // 8 edges (waves) per block
        spmm_scatter<<<blocks, 256, 0, stream>>>(rows, cols, vals, h, out,
                                                 n_edges, n_nodes);
    }

    // K3: clamp logvar half
    {
        long nlv = n_nodes * LATENT;
        clamp_lv<<<(int)((nlv + 255) / 256), 256, 0, stream>>>(out + nlv, nlv);
    }
}